// PointSample_3762391351538
// MI455X (gfx1250) — compile-verified
//
#include <hip/hip_runtime.h>

typedef __attribute__((ext_vector_type(2))) float v2f;
typedef __attribute__((ext_vector_type(8))) float v8f;

#define NXc 128
#define NYc 128
#define NZc 8
#define Pt  (NXc * NYc * NZc)   // 131072 world points
#define Cc  128                 // feature channels
#define Hf  116                 // feature map height
#define Wf  200                 // feature map width
#define HWf (Hf * Wf)           // 23200
#define Vn  6                   // views

// ---------------------------------------------------------------------------
// Kernel 1: projection via V_WMMA_F32_16X16X4_F32.
// Per wave: 16 world points.  A (16x4) = rows of the stacked 24x4 projection
// matrix (proj_mats is [1,6,4,4] -> flat 24 rows x 4 cols).  B (4x16) = the
// 16 points' homogeneous world coords.  D (16x16) = cam components.
//   WMMA #1: rows 0..15  -> views 0..3 (4 comps each)
//   WMMA #2: rows 16..23 -> views 4..5 (rows 8..15 of A zeroed)
// A f32 16x4 layout: lanes 0-15 hold K=0,1 in vgpr0,1; lanes 16-31 hold K=2,3.
// B f32 4x16 layout: vgpr r = row K=r (lanes 0-15) / K=r+2 (lanes 16-31).
// D f32 16x16 layout: vgpr r = row M=r (lanes 0-15) / M=r+8 (lanes 16-31).
// ---------------------------------------------------------------------------
__global__ __launch_bounds__(256) void proj_wmma_kernel(
    const float* __restrict__ proj,     // [6,4,4] flat = 24x4
    float* __restrict__ world_out,      // [P,4]
    float* __restrict__ cam_out)        // [6,P,4]
{
  const int lane = threadIdx.x & 31;
  const int wave = threadIdx.x >> 5;
  const int pointBase = (blockIdx.x * 8 + wave) * 16;

  const int n  = lane & 15;      // point column / A row
  const int hi = lane >> 4;      // half-wave select
  const int kb = hi << 1;        // K base: 0 or 2

  // world coordinates of this lane's point (linspace, align ends)
  const int p  = pointBase + n;
  const int ix = p & (NXc - 1);
  const int iy = (p >> 7) & (NYc - 1);
  const int iz = p >> 14;
  const float wx = -51.2f + (float)ix * (102.4f / 127.0f);
  const float wy = -51.2f + (float)iy * (102.4f / 127.0f);
  const float wz = -5.0f  + (float)iz * (8.0f / 7.0f);

  // B operand: (K=kb, K=kb+1) of column n  -> (x,y) low half, (z,1) high half
  v2f b;
  b.x = hi ? wz   : wx;
  b.y = hi ? 1.0f : wy;

  // A operands: row m = n, cols kb..kb+1 of stacked projection matrix
  v2f a1;
  a1.x = proj[n * 4 + kb];
  a1.y = proj[n * 4 + kb + 1];
  v2f a2 = {0.0f, 0.0f};
  if (n < 8) {                       // rows 16..23 (views 4,5); rest zero
    a2.x = proj[(16 + n) * 4 + kb];
    a2.y = proj[(16 + n) * 4 + kb + 1];
  }

  v8f c0 = {};
  // 8 args: (neg_a, A, neg_b, B, c_mod, C, reuse_a, reuse_b)
  v8f d1 = __builtin_amdgcn_wmma_f32_16x16x4_f32(
      false, a1, false, b, (short)0, c0, false, false);
  v8f d2 = __builtin_amdgcn_wmma_f32_16x16x4_f32(
      false, a2, false, b, (short)0, c0, false, false);

  // scatter D1 -> cam_points for views 0..3
#pragma unroll
  for (int r = 0; r < 8; ++r) {
    const int mr = r + (hi << 3);                // output row 0..15
    const int v  = mr >> 2;
    const int cc = mr & 3;
    cam_out[((size_t)v * Pt + (size_t)p) * 4 + cc] = d1[r];
  }
  // D2 rows 0..7 (lanes 0..15) -> views 4..5; rows 8..15 are zero padding
  if (hi == 0) {
#pragma unroll
    for (int r = 0; r < 8; ++r) {
      const int v  = 4 + (r >> 2);
      const int cc = r & 3;
      cam_out[((size_t)v * Pt + (size_t)p) * 4 + cc] = d2[r];
    }
    float4 wp = make_float4(wx, wy, wz, 1.0f);
    *(float4*)(world_out + (size_t)p * 4) = wp;
  }
}

// ---------------------------------------------------------------------------
// Kernel 2: bilinear sampling + view mean.  One lane per spatial point.
// Per view precompute: clamped corner base offset + 4 fused weights
// (bilinear * valid-mask * 1/6).  x-corner pair is contiguous -> one 8B load
// per row, 2 loads per view per channel (12 gathers/channel instead of 24).
// Whole-wave early-out when every lane has zero total weight.
// Stores bev[c*P + p]: fully coalesced.
// ---------------------------------------------------------------------------
__global__ __launch_bounds__(256) void sample_kernel(
    const float* __restrict__ feat,     // [6,128,116,200]
    const float* __restrict__ cam,      // [6,P,4]
    float* __restrict__ bev)            // [128,P]
{
  const int p = blockIdx.x * blockDim.x + threadIdx.x;

  int   base[Vn];
  float w00[Vn], w01[Vn], w10[Vn], w11[Vn];
  float wsum = 0.0f;

#pragma unroll
  for (int v = 0; v < Vn; ++v) {
    const float4 cp = ((const float4*)cam)[(size_t)v * Pt + p];
    const float depth = fmaxf(cp.z, 1e-4f);
    const float xi = cp.x / depth;
    const float yi = cp.y / depth;
    const bool valid = (xi >= 0.0f) && (xi <= 1599.0f) &&
                       (yi >= 0.0f) && (yi <= 927.0f);
    // image px -> feature px (align_corners=True)
    const float xp = xi * (199.0f / 1599.0f);
    const float yp = yi * (115.0f / 927.0f);
    const float x0f = floorf(xp), y0f = floorf(yp);
    const float wx1 = xp - x0f, wx0 = 1.0f - wx1;
    const float wy1 = yp - y0f, wy0 = 1.0f - wy1;
    const int x0 = (int)x0f, y0 = (int)y0f;
    const int xb = min(max(x0, 0), Wf - 2);
    const int yb = min(max(y0, 0), Hf - 2);
    // remap weights onto clamped pair (xb, xb+1); OOB corners weight 0
    float wxA, wxB, wyA, wyB;
    if (x0 == xb) { wxA = wx0;  wxB = (x0 < Wf - 1) ? wx1 : 0.0f; }
    else          { wxA = 0.0f; wxB = wx0; }   // x0 == Wf-1 (or invalid)
    if (y0 == yb) { wyA = wy0;  wyB = (y0 < Hf - 1) ? wy1 : 0.0f; }
    else          { wyA = 0.0f; wyB = wy0; }
    const float s = valid ? (1.0f / 6.0f) : 0.0f;   // fold mask + view mean
    w00[v] = wxA * wyA * s;  w01[v] = wxB * wyA * s;
    w10[v] = wxA * wyB * s;  w11[v] = wxB * wyB * s;
    base[v] = v * (Cc * HWf) + yb * Wf + xb;
    wsum += w00[v] + w01[v] + w10[v] + w11[v];
  }

  if (__ballot(wsum > 0.0f) == 0ull) {      // whole wave contributes nothing
    for (int c = 0; c < Cc; ++c)
      bev[(size_t)c * Pt + p] = 0.0f;
    return;
  }

  for (int c = 0; c < Cc; ++c) {
    float acc = 0.0f;
#pragma unroll
    for (int v = 0; v < Vn; ++v) {
      const float* f0 = feat + base[v] + c * HWf;
      v2f lo, hv;
      __builtin_memcpy(&lo, f0, 8);        // (x0,y0),(x0+1,y0)
      __builtin_memcpy(&hv, f0 + Wf, 8);   // (x0,y0+1),(x0+1,y0+1)
      acc += lo.x * w00[v] + lo.y * w01[v] + hv.x * w10[v] + hv.y * w11[v];
    }
    bev[(size_t)c * Pt + p] = acc;
  }
}

// ---------------------------------------------------------------------------
// d_out layout (flat, return order):
//   bev_feats  [1,128,8,128,128] : 16,777,216 floats @ 0
//   world_pts  [1,8,128,128,4]   :    524,288 floats @ 16,777,216
//   cam_points [1,6,8,128,128,4] :  3,145,728 floats @ 17,301,504
// ---------------------------------------------------------------------------
extern "C" void kernel_launch(void* const* d_in, const int* in_sizes, int n_in,
                              void* d_out, int out_size, void* d_ws, size_t ws_size,
                              hipStream_t stream) {
  (void)in_sizes; (void)n_in; (void)out_size; (void)d_ws; (void)ws_size;
  const float* feat = (const float*)d_in[0];   // [6,128,116,200]
  const float* proj = (const float*)d_in[1];   // [1,6,4,4]

  float* out   = (float*)d_out;
  float* bev   = out;                                  // 16,777,216
  float* world = out + (size_t)16777216;               //    524,288
  float* camp  = out + (size_t)16777216 + 524288;      //  3,145,728

  // 1024 blocks * 8 waves * 16 points = 131072 points
  proj_wmma_kernel<<<Pt / 128, 256, 0, stream>>>(proj, world, camp);
  // 512 blocks * 256 lanes = 131072 points
  sample_kernel<<<Pt / 256, 256, 0, stream>>>(feat, camp, bev);
}